// DeepseekOcrTextMoe_75093208203453
// MI455X (gfx1250) — compile-verified
//
#include <hip/hip_runtime.h>

// ---------------- problem constants (match reference) ----------------
#define BD   2            // batch
#define SD   1024         // seq
#define DD   2048         // hidden
#define EE   32           // routed experts
#define HH   1408         // routed intermediate
#define KK   6            // top-k
#define HS   2816         // shared intermediate
#define TT   (BD*SD)      // tokens = 2048
#define RSCALE 1.0f

#define ROWS_PER_TILE 32
#define MAX_TILES (TT*KK/ROWS_PER_TILE + EE)   // 416: upper bound on padded 32-row tiles

typedef __attribute__((ext_vector_type(2))) float v2f;
typedef __attribute__((ext_vector_type(8))) float v8f;

static __device__ __forceinline__ v8f wmma_f32(v2f a, v2f b, v8f c) {
  // D = A(16x4 f32) * B(4x16 f32) + C(16x16 f32)
  return __builtin_amdgcn_wmma_f32_16x16x4_f32(
      /*neg_a=*/false, a, /*neg_b=*/false, b,
      /*c_mod=*/(short)0, c, /*reuse_a=*/false, /*reuse_b=*/false);
}

static __device__ __forceinline__ float silu(float g) {
  // g * rcp(1+exp(-g)): v_exp_f32 + v_rcp_f32, no IEEE-div fixup chain
  return g * __builtin_amdgcn_rcpf(1.0f + __expf(-g));
}

// ---------------- 1) router: softmax + greedy top-6 -----------------
// one wave32 per token; lane e owns expert e (E==32 fits wave32 exactly)
__global__ __launch_bounds__(32) void moe_router_kernel(
    const float* __restrict__ x, const float* __restrict__ gate_w,
    int* __restrict__ topk_i, float* __restrict__ topk_w) {
  const int t = blockIdx.x;
  const int e = threadIdx.x;
  const float* xr = x + (size_t)t * DD;
  const float* gr = gate_w + (size_t)e * DD;
  float acc = 0.0f;
  for (int d = 0; d < DD; d += 4) {
    float4 xv = *(const float4*)(xr + d);
    float4 gv = *(const float4*)(gr + d);
    acc = fmaf(xv.x, gv.x, fmaf(xv.y, gv.y, fmaf(xv.z, gv.z, fmaf(xv.w, gv.w, acc))));
  }
  // wave32 softmax across lanes
  float m = acc;
  for (int off = 16; off; off >>= 1) m = fmaxf(m, __shfl_xor(m, off, 32));
  float p = __expf(acc - m);
  float s = p;
  for (int off = 16; off; off >>= 1) s += __shfl_xor(s, off, 32);
  float score = p / s;
  // greedy top-K, lower index wins ties (top_k semantics)
  for (int k = 0; k < KK; ++k) {
    float v = score; int i = e;
    for (int off = 16; off; off >>= 1) {
      float ov = __shfl_xor(v, off, 32);
      int   oi = __shfl_xor(i, off, 32);
      if (ov > v || (ov == v && oi < i)) { v = ov; i = oi; }
    }
    if (e == i) score = -1.0f;
    if (e == k) { topk_i[t*KK + k] = i; topk_w[t*KK + k] = v * RSCALE; }
  }
}

// -------- 2) deterministic per-expert token lists + prefix offsets --------
// also builds tok2slot: (token,k) -> global row slot in dbuf/hbuf
__global__ __launch_bounds__(32) void moe_build_lists_kernel(
    const int* __restrict__ topk_i, const float* __restrict__ topk_w,
    int* __restrict__ counts, int* __restrict__ rowOff, int* __restrict__ tileOff,
    int* __restrict__ totalTiles, int* __restrict__ list,
    int* __restrict__ tkidx, int* __restrict__ tok2slot) {
  const int e = threadIdx.x;            // one lane per expert
  int cnt = 0;
  for (int t = 0; t < TT; ++t)
    for (int k = 0; k < KK; ++k)
      if (topk_i[t*KK + k] == e) {
        list [e*TT + cnt] = t;
        tkidx[e*TT + cnt] = t*KK + k;
        ++cnt;
      }
  counts[e] = cnt;
  __syncthreads();
  if (e == 0) {
    int ro = 0, to = 0;
    for (int j = 0; j < EE; ++j) {
      rowOff[j] = ro; tileOff[j] = to;
      ro += counts[j];
      to += (counts[j] + ROWS_PER_TILE - 1) / ROWS_PER_TILE;
    }
    rowOff[EE] = ro; tileOff[EE] = to;
    *totalTiles = to;
  }
  __syncthreads();
  const int base = rowOff[e];
  for (int i = 0; i < cnt; ++i)
    tok2slot[tkidx[e*TT + i]] = base + i;
  (void)topk_w;
}

// -------- 3a) shared expert: hs = silu(x@swg) * (x@swu) --------
// one wave per 32x32 tile: 2 M-tiles x 2 N-tiles x {gate,up} = 8 accumulators.
// per k-substep: 2 b64 A loads + 8 b32 B loads feed 8 WMMAs.
__global__ __launch_bounds__(32) void moe_shared_gateup_kernel(
    const float* __restrict__ x, const float* __restrict__ swg,
    const float* __restrict__ swu, float* __restrict__ hs) {
  const int t0   = blockIdx.x * 32;
  const int c0   = blockIdx.y * 32;
  const int lane = threadIdx.x;
  const int m    = lane & 15;
  const int koff = (lane >> 4) * 2;          // A/B K-pair select per ISA layout
  const int col0 = c0 + m, col1 = c0 + 16 + m;
  const float* xr0 = x + (size_t)(t0 + m) * DD;
  const float* xr1 = x + (size_t)(t0 + 16 + m) * DD;
  __builtin_prefetch(swg + col0, 0, 1);      // global_prefetch_b8 on B streams
  __builtin_prefetch(swu + col0, 0, 1);
  v8f cg[2][2] = {}; v8f cu[2][2] = {};
  for (int k = 0; k < DD; k += 4) {
    float2 av0 = *(const float2*)(xr0 + k + koff);
    float2 av1 = *(const float2*)(xr1 + k + koff);
    v2f a0; a0.x = av0.x; a0.y = av0.y;
    v2f a1; a1.x = av1.x; a1.y = av1.y;
    const size_t bi = (size_t)(k + koff) * HS;
    v2f bg0; bg0.x = swg[bi + col0]; bg0.y = swg[bi + HS + col0];
    v2f bg1; bg1.x = swg[bi + col1]; bg1.y = swg[bi + HS + col1];
    v2f bu0; bu0.x = swu[bi + col0]; bu0.y = swu[bi + HS + col0];
    v2f bu1; bu1.x = swu[bi + col1]; bu1.y = swu[bi + HS + col1];
    cg[0][0] = wmma_f32(a0, bg0, cg[0][0]); cg[0][1] = wmma_f32(a0, bg1, cg[0][1]);
    cg[1][0] = wmma_f32(a1, bg0, cg[1][0]); cg[1][1] = wmma_f32(a1, bg1, cg[1][1]);
    cu[0][0] = wmma_f32(a0, bu0, cu[0][0]); cu[0][1] = wmma_f32(a0, bu1, cu[0][1]);
    cu[1][0] = wmma_f32(a1, bu0, cu[1][0]); cu[1][1] = wmma_f32(a1, bu1, cu[1][1]);
  }
  const int rbase = (lane >> 4) * 8;          // C layout: M = r + 8*(lane>=16)
  #pragma unroll
  for (int ti = 0; ti < 2; ++ti)
    #pragma unroll
    for (int r = 0; r < 8; ++r) {
      const size_t row = (size_t)(t0 + ti*16 + rbase + r);
      hs[row * HS + col0] = silu(cg[ti][0][r]) * cu[ti][0][r];
      hs[row * HS + col1] = silu(cg[ti][1][r]) * cu[ti][1][r];
    }
}

// -------- 3b) shared expert down-proj: out = hs @ swd (initializes out) --------
// one wave per 32x64 tile: 2 M-tiles x 4 N-tiles = 8 accumulators.
__global__ __launch_bounds__(32) void moe_shared_down_kernel(
    const float* __restrict__ hs, const float* __restrict__ swd,
    float* __restrict__ out) {
  const int t0   = blockIdx.x * 32;
  const int c0   = blockIdx.y * 64;
  const int lane = threadIdx.x;
  const int m    = lane & 15;
  const int koff = (lane >> 4) * 2;
  const float* hr0 = hs + (size_t)(t0 + m) * HS;
  const float* hr1 = hs + (size_t)(t0 + 16 + m) * HS;
  __builtin_prefetch(swd + c0 + m, 0, 1);
  v8f c[2][4] = {};
  for (int k = 0; k < HS; k += 4) {
    float2 av0 = *(const float2*)(hr0 + k + koff);
    float2 av1 = *(const float2*)(hr1 + k + koff);
    v2f a0; a0.x = av0.x; a0.y = av0.y;
    v2f a1; a1.x = av1.x; a1.y = av1.y;
    const size_t bi = (size_t)(k + koff) * DD;
    v2f b[4];
    #pragma unroll
    for (int tj = 0; tj < 4; ++tj) {
      const int col = c0 + tj*16 + m;
      b[tj].x = swd[bi + col]; b[tj].y = swd[bi + DD + col];
    }
    #pragma unroll
    for (int tj = 0; tj < 4; ++tj) {
      c[0][tj] = wmma_f32(a0, b[tj], c[0][tj]);
      c[1][tj] = wmma_f32(a1, b[tj], c[1][tj]);
    }
  }
  const int rbase = (lane >> 4) * 8;
  #pragma unroll
  for (int ti = 0; ti < 2; ++ti)
    #pragma unroll
    for (int tj = 0; tj < 4; ++tj)
      #pragma unroll
      for (int r = 0; r < 8; ++r)
        out[(size_t)(t0 + ti*16 + rbase + r) * DD + (c0 + tj*16 + m)] = c[ti][tj][r];
}

// -------- 4a) routed experts grouped GEMM: h = silu(Xg@wg_e)*(Xg@wu_e) --------
// 32 gathered rows x 32 cols per wave; grid.x = padded 32-row tiles (early-exit).
__global__ __launch_bounds__(32) void moe_routed_gateup_kernel(
    const float* __restrict__ x, const float* __restrict__ wg,
    const float* __restrict__ wu,
    const int* __restrict__ tileOff, const int* __restrict__ rowOff,
    const int* __restrict__ counts, const int* __restrict__ list,
    const int* __restrict__ totalTiles, float* __restrict__ hbuf) {
  const int bt = blockIdx.x;
  if (bt >= *totalTiles) return;               // uniform exit
  int e = 0;
  while (tileOff[e + 1] <= bt) ++e;            // E=32, uniform linear search
  const int lt   = bt - tileOff[e];
  const int cnt  = counts[e];
  const int lane = threadIdx.x;
  const int m    = lane & 15;
  const int koff = (lane >> 4) * 2;
  const int li0  = lt * 32 + m;
  const int li1  = li0 + 16;
  const int row0 = (li0 < cnt) ? list[e * TT + li0] : 0;   // pad rows -> token 0
  const int row1 = (li1 < cnt) ? list[e * TT + li1] : 0;
  const int c0   = blockIdx.y * 32;
  const int col0 = c0 + m, col1 = c0 + 16 + m;
  const float* xr0 = x + (size_t)row0 * DD;
  const float* xr1 = x + (size_t)row1 * DD;
  const float* wge = wg + (size_t)e * DD * HH;
  const float* wue = wu + (size_t)e * DD * HH;
  __builtin_prefetch(wge + col0, 0, 1);
  __builtin_prefetch(wue + col0, 0, 1);
  v8f cg[2][2] = {}; v8f cu[2][2] = {};
  for (int k = 0; k < DD; k += 4) {
    float2 av0 = *(const float2*)(xr0 + k + koff);
    float2 av1 = *(const float2*)(xr1 + k + koff);
    v2f a0; a0.x = av0.x; a0.y = av0.y;
    v2f a1; a1.x = av1.x; a1.y = av1.y;
    const size_t bi = (size_t)(k + koff) * HH;
    v2f bg0; bg0.x = wge[bi + col0]; bg0.y = wge[bi + HH + col0];
    v2f bg1; bg1.x = wge[bi + col1]; bg1.y = wge[bi + HH + col1];
    v2f bu0; bu0.x = wue[bi + col0]; bu0.y = wue[bi + HH + col0];
    v2f bu1; bu1.x = wue[bi + col1]; bu1.y = wue[bi + HH + col1];
    cg[0][0] = wmma_f32(a0, bg0, cg[0][0]); cg[0][1] = wmma_f32(a0, bg1, cg[0][1]);
    cg[1][0] = wmma_f32(a1, bg0, cg[1][0]); cg[1][1] = wmma_f32(a1, bg1, cg[1][1]);
    cu[0][0] = wmma_f32(a0, bu0, cu[0][0]); cu[0][1] = wmma_f32(a0, bu1, cu[0][1]);
    cu[1][0] = wmma_f32(a1, bu0, cu[1][0]); cu[1][1] = wmma_f32(a1, bu1, cu[1][1]);
  }
  const int slot0 = rowOff[e] + lt * 32;
  const int rbase = (lane >> 4) * 8;
  #pragma unroll
  for (int ti = 0; ti < 2; ++ti)
    #pragma unroll
    for (int r = 0; r < 8; ++r) {
      const int lr = lt * 32 + ti*16 + rbase + r;
      if (lr < cnt) {                            // don't spill into next expert
        const size_t row = (size_t)(slot0 + ti*16 + rbase + r);
        hbuf[row * HH + col0] = silu(cg[ti][0][r]) * cu[ti][0][r];
        hbuf[row * HH + col1] = silu(cg[ti][1][r]) * cu[ti][1][r];
      }
    }
}

// -------- 4b) routed down-proj: dbuf[slot] = h[slot] @ wd_e (unscaled) --------
// 32 rows x 64 cols per wave; plain stores, no atomics.
__global__ __launch_bounds__(32) void moe_routed_down_kernel(
    const float* __restrict__ hbuf, const float* __restrict__ wd,
    const int* __restrict__ tileOff, const int* __restrict__ rowOff,
    const int* __restrict__ counts, const int* __restrict__ totalTiles,
    float* __restrict__ dbuf) {
  const int bt = blockIdx.x;
  if (bt >= *totalTiles) return;
  int e = 0;
  while (tileOff[e + 1] <= bt) ++e;
  const int lt   = bt - tileOff[e];
  const int cnt  = counts[e];
  const int lane = threadIdx.x;
  const int m    = lane & 15;
  const int koff = (lane >> 4) * 2;
  const int c0   = blockIdx.y * 64;
  const int slot0 = rowOff[e] + lt * 32;
  const float* hr0 = hbuf + (size_t)(slot0 + m) * HH;       // tail padded in ws
  const float* hr1 = hbuf + (size_t)(slot0 + 16 + m) * HH;
  const float* wde = wd + (size_t)e * HH * DD;
  __builtin_prefetch(wde + c0 + m, 0, 1);
  v8f c[2][4] = {};
  for (int k = 0; k < HH; k += 4) {
    float2 av0 = *(const float2*)(hr0 + k + koff);
    float2 av1 = *(const float2*)(hr1 + k + koff);
    v2f a0; a0.x = av0.x; a0.y = av0.y;
    v2f a1; a1.x = av1.x; a1.y = av1.y;
    const size_t bi = (size_t)(k + koff) * DD;
    v2f b[4];
    #pragma unroll
    for (int tj = 0; tj < 4; ++tj) {
      const int col = c0 + tj*16 + m;
      b[tj].x = wde[bi + col]; b[tj].y = wde[bi + DD + col];
    }
    #pragma unroll
    for (int tj = 0; tj < 4; ++tj) {
      c[0][tj] = wmma_f32(a0, b[tj], c[0][tj]);
      c[1][tj] = wmma_f32(a1, b[tj], c[1][tj]);
    }
  }
  const int rbase = (lane >> 4) * 8;
  #pragma unroll
  for (int ti = 0; ti < 2; ++ti)
    #pragma unroll
    for (int r = 0; r < 8; ++r) {
      const int lr = lt * 32 + ti*16 + rbase + r;
      if (lr < cnt) {
        const size_t row = (size_t)(slot0 + ti*16 + rbase + r);
        #pragma unroll
        for (int tj = 0; tj < 4; ++tj)
          dbuf[row * DD + (c0 + tj*16 + m)] = c[ti][tj][r];
      }
    }
}

// -------- 5) combine: out[t] = shared[t] + sum_k w_k * dbuf[slot_k]  --------
// fixed k-order per token -> bitwise deterministic; coalesced b128 traffic.
__global__ __launch_bounds__(256) void moe_combine_kernel(
    const float* __restrict__ dbuf, const int* __restrict__ tok2slot,
    const float* __restrict__ topk_w, float* __restrict__ out) {
  const int t  = blockIdx.x;
  const int d0 = threadIdx.x * 8;                 // 256 threads * 8 = 2048 = D
  float* op = out + (size_t)t * DD + d0;
  float4 acc0 = *(const float4*)(op);
  float4 acc1 = *(const float4*)(op + 4);
  #pragma unroll
  for (int k = 0; k < KK; ++k) {
    const int   slot = tok2slot[t*KK + k];        // block-uniform -> SMEM loads
    const float w    = topk_w [t*KK + k];
    const float* dp  = dbuf + (size_t)slot * DD + d0;
    const float4 v0 = *(const float4*)(dp);
    const float4 v1 = *(const float4*)(dp + 4);
    acc0.x = fmaf(w, v0.x, acc0.x); acc0.y = fmaf(w, v0.y, acc0.y);
    acc0.z = fmaf(w, v0.z, acc0.z); acc0.w = fmaf(w, v0.w, acc0.w);
    acc1.x = fmaf(w, v1.x, acc1.x); acc1.y = fmaf(w, v1.y, acc1.y);
    acc1.z = fmaf(w, v1.z, acc1.z); acc1.w = fmaf(w, v1.w, acc1.w);
  }
  *(float4*)(op)     = acc0;
  *(float4*)(op + 4) = acc1;
}

// ---------------- host-side launch ----------------
extern "C" void kernel_launch(void* const* d_in, const int* in_sizes, int n_in,
                              void* d_out, int out_size, void* d_ws, size_t ws_size,
                              hipStream_t stream) {
  (void)in_sizes; (void)n_in; (void)out_size; (void)ws_size;
  const float* x      = (const float*)d_in[0];   // [B,S,D]
  const float* gate_w = (const float*)d_in[1];   // [E,D]
  const float* wg     = (const float*)d_in[2];   // [E,D,H]
  const float* wu     = (const float*)d_in[3];   // [E,D,H]
  const float* wd     = (const float*)d_in[4];   // [E,H,D]
  const float* swg    = (const float*)d_in[5];   // [D,HS]
  const float* swu    = (const float*)d_in[6];   // [D,HS]
  const float* swd    = (const float*)d_in[7];   // [HS,D]
  float* out = (float*)d_out;

  // workspace carve-up (256B aligned)
  char* ws = (char*)d_ws;
  size_t off = 0;
  auto take = [&](size_t bytes) { char* p = ws + off; off = (off + bytes + 255) & ~(size_t)255; return p; };
  int*   topk_i     = (int*)  take((size_t)TT * KK * 4);
  float* topk_w     = (float*)take((size_t)TT * KK * 4);
  int*   counts     = (int*)  take((size_t)EE * 4);
  int*   rowOff     = (int*)  take((size_t)(EE + 1) * 4);
  int*   tileOff    = (int*)  take((size_t)(EE + 1) * 4);
  int*   totalTiles = (int*)  take(4);
  int*   list       = (int*)  take((size_t)EE * TT * 4);
  int*   tkidx      = (int*)  take((size_t)EE * TT * 4);
  int*   tok2slot   = (int*)  take((size_t)TT * KK * 4);
  float* hs         = (float*)take((size_t)TT * HS * 4);
  float* hbuf       = (float*)take(((size_t)TT * KK + ROWS_PER_TILE) * HH * 4); // +32 row pad
  float* dbuf       = (float*)take((size_t)TT * KK * DD * 4);

  // 1) router (T waves)
  moe_router_kernel<<<TT, 32, 0, stream>>>(x, gate_w, topk_i, topk_w);
  // 2) deterministic per-expert lists + offsets + token->slot map
  moe_build_lists_kernel<<<1, 32, 0, stream>>>(topk_i, topk_w, counts, rowOff,
                                               tileOff, totalTiles, list, tkidx, tok2slot);
  // 3) shared expert
  moe_shared_gateup_kernel<<<dim3(TT / 32, HS / 32), 32, 0, stream>>>(x, swg, swu, hs);
  moe_shared_down_kernel<<<dim3(TT / 32, DD / 64), 32, 0, stream>>>(hs, swd, out);
  // 4) routed experts (grouped GEMMs over gathered token tiles)
  moe_routed_gateup_kernel<<<dim3(MAX_TILES, HH / 32), 32, 0, stream>>>(
      x, wg, wu, tileOff, rowOff, counts, list, totalTiles, hbuf);
  moe_routed_down_kernel<<<dim3(MAX_TILES, DD / 64), 32, 0, stream>>>(
      hbuf, wd, tileOff, rowOff, counts, totalTiles, dbuf);
  // 5) deterministic weighted combine (fixed k-order per token)
  moe_combine_kernel<<<TT, 256, 0, stream>>>(dbuf, tok2slot, topk_w, out);
}